// RelationalAttentionProb_64991445123873
// MI455X (gfx1250) — compile-verified
//
#include <hip/hip_runtime.h>

// RelationalAttentionProb on MI455X (gfx1250, wave32).
//
// Stage 1: T = x @ w_src, U = x @ w_dst via V_WMMA_F32_16X16X4_F32 (exact f32).
//          One wave per 16-node tile; N = 16 relations = one WMMA tile;
//          K = 128 -> 32 WMMA ops per accumulator, 2 accumulators per wave.
// Stage 2: per-edge gather from the 6.4MB tables (L2-resident) + sigmoid/clamp.

#define NUM_NODES     50000
#define INPUT_SIZE    128
#define NUM_RELATIONS 16
#define NUM_EDGES     1600000
#define NODE_TILES    (NUM_NODES / 16)   // 3125, exact
#define WAVES_PER_BLK 8                  // 256 threads = 8 wave32

typedef __attribute__((ext_vector_type(2))) float v2f;
typedef __attribute__((ext_vector_type(8))) float v8f;

__global__ __launch_bounds__(256) void node_logits_wmma_kernel(
    const float* __restrict__ x,          // [NUM_NODES][INPUT_SIZE]
    const float* __restrict__ att_weight, // [2*INPUT_SIZE][NUM_RELATIONS]
    float* __restrict__ T,                // [NUM_NODES][NUM_RELATIONS]
    float* __restrict__ U)                // [NUM_NODES][NUM_RELATIONS]
{
    // Stage the whole 16KB weight matrix in LDS once per block.
    __shared__ float wlds[2 * INPUT_SIZE * NUM_RELATIONS];
    for (int i = threadIdx.x; i < 2 * INPUT_SIZE * NUM_RELATIONS; i += 256)
        wlds[i] = att_weight[i];
    __syncthreads();

    const int wave = threadIdx.x >> 5;
    const int lane = threadIdx.x & 31;
    const int tile = blockIdx.x * WAVES_PER_BLK + wave;
    if (tile >= NODE_TILES) return;

    const int half = lane >> 4;   // 0: K pair {0,1} / M 0..7 ; 1: K pair {2,3} / M 8..15
    const int l15  = lane & 15;   // A: row M ; B/C/D: col N
    const int koff = half * 2;

    const float* xrow = x + (size_t)(tile * 16 + l15) * INPUT_SIZE;

    v8f cS = {};  // accumulates x @ w_src tile
    v8f cD = {};  // accumulates x @ w_dst tile

    #pragma unroll 4
    for (int k0 = 0; k0 < INPUT_SIZE; k0 += 4) {
        // A fragment 16x4 (ISA 7.12.2): lane l15 = row M, VGPR0/1 = K koff/koff+1
        v2f a;
        a.x = xrow[k0 + koff + 0];
        a.y = xrow[k0 + koff + 1];
        // B fragments 4x16: lane l15 = col N, VGPR0/1 = K koff/koff+1
        v2f bS, bD;
        bS.x = wlds[(k0 + koff + 0) * NUM_RELATIONS + l15];
        bS.y = wlds[(k0 + koff + 1) * NUM_RELATIONS + l15];
        bD.x = wlds[(INPUT_SIZE + k0 + koff + 0) * NUM_RELATIONS + l15];
        bD.y = wlds[(INPUT_SIZE + k0 + koff + 1) * NUM_RELATIONS + l15];

        // 8-arg form: (neg_a, A, neg_b, B, c_mod, C, reuse_a, reuse_b)
        cS = __builtin_amdgcn_wmma_f32_16x16x4_f32(false, a, false, bS,
                                                   (short)0, cS, false, false);
        cD = __builtin_amdgcn_wmma_f32_16x16x4_f32(false, a, false, bD,
                                                   (short)0, cD, false, false);
    }

    // C/D layout: lanes 0-15 hold M = j, lanes 16-31 hold M = j + 8; N = l15.
    float* Tout = T + (size_t)tile * 16 * NUM_RELATIONS;
    float* Uout = U + (size_t)tile * 16 * NUM_RELATIONS;
    #pragma unroll
    for (int j = 0; j < 8; ++j) {
        const int m = j + half * 8;
        Tout[m * NUM_RELATIONS + l15] = cS[j];
        Uout[m * NUM_RELATIONS + l15] = cD[j];
    }
}

__global__ __launch_bounds__(256) void edge_prob_kernel(
    const int* __restrict__ edge_index,   // [2][NUM_EDGES] (int32: JAX x64 off)
    const int* __restrict__ edge_type,    // [NUM_EDGES]
    const float* __restrict__ T,
    const float* __restrict__ U,
    float* __restrict__ out)
{
    const int e = blockIdx.x * 256 + threadIdx.x;
    if (e >= NUM_EDGES) return;

    const int s = edge_index[e];
    const int d = edge_index[NUM_EDGES + e];
    const int r = edge_type[e];

    const float logit = T[s * NUM_RELATIONS + r] + U[d * NUM_RELATIONS + r];
    float att = 1.0f / (1.0f + __expf(-logit));   // sigmoid via v_exp_f32
    att = fminf(fmaxf(att, 1e-5f), 0.99999f);
    out[e] = att;
}

extern "C" void kernel_launch(void* const* d_in, const int* in_sizes, int n_in,
                              void* d_out, int out_size, void* d_ws, size_t ws_size,
                              hipStream_t stream) {
    const float* x  = (const float*)d_in[0];
    const float* w  = (const float*)d_in[1];
    const int*   ei = (const int*)d_in[2];
    const int*   et = (const int*)d_in[3];
    float* out = (float*)d_out;

    // Workspace: two [NUM_NODES][NUM_RELATIONS] f32 tables = 6.4 MB,
    // fully rewritten by stage 1 every call (poison-safe, deterministic).
    float* T = (float*)d_ws;
    float* U = T + (size_t)NUM_NODES * NUM_RELATIONS;

    const int blocks1 = (NODE_TILES + WAVES_PER_BLK - 1) / WAVES_PER_BLK; // 391
    node_logits_wmma_kernel<<<blocks1, 256, 0, stream>>>(x, w, T, U);

    const int blocks2 = (NUM_EDGES + 255) / 256;
    edge_prob_kernel<<<blocks2, 256, 0, stream>>>(ei, et, T, U, out);
}